// MyGCNII_70042326663944
// MI455X (gfx1250) — compile-verified
//
#include <hip/hip_runtime.h>
#include <hip/hip_bf16.h>
#include <cmath>

// ---------------------------------------------------------------------------
// GCNII forward for MI455X (gfx1250, wave32).
// GEMMs use V_WMMA_F32_16X16X4_F32 (one wave -> one 16x16 f32 tile).
// Sparse aggregation: wave-per-edge gather + global_atomic_add_f32 scatter.
// ---------------------------------------------------------------------------

#define ALPHA 0.1f
#define HID 64
#define IN_F 128
#define OUT_F 10
#define NGRAPH 256

typedef float v2f __attribute__((ext_vector_type(2)));
typedef float v8f __attribute__((ext_vector_type(8)));

// ---------------- degree / norm -------------------------------------------

__global__ void k_init_deg(float* deg, int n) {
  int i = blockIdx.x * blockDim.x + threadIdx.x;
  if (i < n) deg[i] = 1.0f;  // self-loop contributes 1
}

__global__ void k_count_deg(const int* __restrict__ col, float* deg, int E) {
  int e = blockIdx.x * blockDim.x + threadIdx.x;
  if (e < E) atomicAdd(&deg[col[e]], 1.0f);
}

__global__ void k_dinv(float* deg, int n) {
  int i = blockIdx.x * blockDim.x + threadIdx.x;
  if (i < n) deg[i] = rsqrtf(deg[i]);  // deg >= 1 always (self-loops)
}

__global__ void k_norm(const int* __restrict__ row, const int* __restrict__ col,
                       const float* __restrict__ dinv, float* __restrict__ norm,
                       int E, int n) {
  int e = blockIdx.x * blockDim.x + threadIdx.x;
  if (e < E) {
    norm[e] = dinv[row[e]] * dinv[col[e]];
  } else if (e < E + n) {
    float d = dinv[e - E];
    norm[e] = d * d;
  }
}

// ---------------- WMMA GEMM: h0 = relu(X @ W1 + b1) -----------------------
// X:[n,128]  W:[128,64]  H:[n,64].  One wave per 16x16 tile, K-loop step 4.

__global__ __launch_bounds__(32)
void k_gemm1(const float* __restrict__ X, const float* __restrict__ W,
             const float* __restrict__ bias, float* __restrict__ H, int n) {
  const int lane = threadIdx.x & 31;
  const int half = lane >> 4;      // 0: K pair {0,1}, 1: K pair {2,3}
  const int l    = lane & 15;
  const long Mbase = (long)blockIdx.x * 16;
  const int  Nbase = blockIdx.y * 16;

  long arow = Mbase + l;
  if (arow > n - 1) arow = n - 1;  // defensive clamp (n is a multiple of 16)
  const float* xp = X + arow * IN_F + 2 * half;
  const float* wp = W + Nbase + l;

  v8f c = {};
  #pragma unroll
  for (int k = 0; k < IN_F; k += 4) {
    v2f a = *(const v2f*)(xp + k);
    v2f b;
    b.x = wp[(k + 2 * half + 0) * HID];
    b.y = wp[(k + 2 * half + 1) * HID];
    c = __builtin_amdgcn_wmma_f32_16x16x4_f32(false, a, false, b,
                                              (short)0, c, false, false);
  }

  const float bb = bias[Nbase + l];
  #pragma unroll
  for (int v = 0; v < 8; ++v) {
    long r = Mbase + v + 8 * half;
    if (r < n) H[r * HID + Nbase + l] = fmaxf(c[v] + bb, 0.0f);
  }
}

// ---------------- per-layer: s = alpha*h0 ---------------------------------

__global__ void k_init_s(const float* __restrict__ h0, float* __restrict__ s,
                         long total) {
  long i = (long)blockIdx.x * blockDim.x + threadIdx.x;
  if (i < total) s[i] = ALPHA * h0[i];
}

// ---------------- scatter: s[col] += 0.9*norm*h[row] ----------------------
// One wave32 per (edge or self-loop); 64 features = 2 floats per lane.

__global__ void k_scatter(const int* __restrict__ row, const int* __restrict__ col,
                          const float* __restrict__ norm, const float* __restrict__ h,
                          float* __restrict__ s, int E, int n) {
  long gid  = (long)blockIdx.x * blockDim.x + threadIdx.x;
  long wid  = gid >> 5;
  int  lane = (int)(gid & 31);
  if (wid >= (long)E + n) return;
  int r, ctgt;
  if (wid < E) { r = row[wid]; ctgt = col[wid]; }
  else         { r = ctgt = (int)(wid - E); }
  const float w = (1.0f - ALPHA) * norm[wid];
  const float* hr = h + (long)r * HID;
  float*       sc = s + (long)ctgt * HID;
  atomicAdd(&sc[lane],      w * hr[lane]);
  atomicAdd(&sc[lane + 32], w * hr[lane + 32]);
}

// ---------------- WMMA layer GEMM: h = relu((1-b)*s + b*(s@Wl)) -----------

__global__ __launch_bounds__(32)
void k_gemm_layer(const float* __restrict__ S, const float* __restrict__ W,
                  float* __restrict__ H, float beta, int n) {
  const int lane = threadIdx.x & 31;
  const int half = lane >> 4;
  const int l    = lane & 15;
  const long Mbase = (long)blockIdx.x * 16;
  const int  Nbase = blockIdx.y * 16;

  long arow = Mbase + l;
  if (arow > n - 1) arow = n - 1;
  const float* sp = S + arow * HID + 2 * half;
  const float* wp = W + Nbase + l;

  v8f c = {};
  #pragma unroll
  for (int k = 0; k < HID; k += 4) {
    v2f a = *(const v2f*)(sp + k);
    v2f b;
    b.x = wp[(k + 2 * half + 0) * HID];
    b.y = wp[(k + 2 * half + 1) * HID];
    c = __builtin_amdgcn_wmma_f32_16x16x4_f32(false, a, false, b,
                                              (short)0, c, false, false);
  }

  #pragma unroll
  for (int v = 0; v < 8; ++v) {
    long r = Mbase + v + 8 * half;
    if (r < n) {
      float sv = S[r * HID + Nbase + l];
      H[r * HID + Nbase + l] = fmaxf((1.0f - beta) * sv + beta * c[v], 0.0f);
    }
  }
}

// ---------------- pooling --------------------------------------------------

__global__ void k_pool_zero(float* pooled, float* counts) {
  int i = blockIdx.x * blockDim.x + threadIdx.x;
  if (i < NGRAPH * HID) pooled[i] = 0.0f;
  else if (i < NGRAPH * HID + NGRAPH) counts[i - NGRAPH * HID] = 0.0f;
}

__global__ void k_pool(const int* __restrict__ batch, const float* __restrict__ h,
                       float* __restrict__ pooled, float* __restrict__ counts, int n) {
  long gid  = (long)blockIdx.x * blockDim.x + threadIdx.x;
  long wid  = gid >> 5;
  int  lane = (int)(gid & 31);
  if (wid >= n) return;
  int g = batch[wid];
  const float* hr = h + wid * HID;
  atomicAdd(&pooled[g * HID + lane],      hr[lane]);
  atomicAdd(&pooled[g * HID + lane + 32], hr[lane + 32]);
  if (lane == 0) atomicAdd(&counts[g], 1.0f);
}

__global__ void k_final(const float* __restrict__ pooled, const float* __restrict__ counts,
                        const float* __restrict__ w, const float* __restrict__ b,
                        float* __restrict__ out) {
  int i = blockIdx.x * blockDim.x + threadIdx.x;
  if (i >= NGRAPH * OUT_F) return;
  int g = i / OUT_F, o = i % OUT_F;
  float inv = 1.0f / fmaxf(counts[g], 1.0f);
  float acc = b[o];
  #pragma unroll 8
  for (int f = 0; f < HID; ++f) acc += (pooled[g * HID + f] * inv) * w[f * OUT_F + o];
  out[i] = acc;
}

// ---------------- launch ---------------------------------------------------

extern "C" void kernel_launch(void* const* d_in, const int* in_sizes, int n_in,
                              void* d_out, int out_size, void* d_ws, size_t ws_size,
                              hipStream_t stream) {
  const float* x      = (const float*)d_in[0];
  const int*   eidx   = (const int*)  d_in[1];   // [2,E] flat
  const int*   batch  = (const int*)  d_in[2];
  const float* lin1_w = (const float*)d_in[3];
  const float* lin1_b = (const float*)d_in[4];
  const float* convw  = (const float*)d_in[5];   // [3,64,64]
  const float* lin2_w = (const float*)d_in[6];
  const float* lin2_b = (const float*)d_in[7];
  float* out = (float*)d_out;

  const int n = in_sizes[0] / IN_F;      // 50000
  const int E = in_sizes[1] / 2;         // 800000
  const int nlayers = in_sizes[5] / (HID * HID);

  const int* row = eidx;
  const int* col = eidx + E;

  // workspace layout (floats)
  float* ws     = (float*)d_ws;
  float* dinv   = ws;                       // n
  float* norm   = dinv + n;                 // E + n
  float* h0     = norm + (E + n);           // n*HID
  float* h      = h0 + (long)n * HID;       // n*HID
  float* s      = h  + (long)n * HID;       // n*HID
  float* pooled = s  + (long)n * HID;       // NGRAPH*HID
  float* counts = pooled + NGRAPH * HID;    // NGRAPH

  const int B = 256;
  const long nh = (long)n * HID;

  // 1) symmetric normalization coefficients
  k_init_deg<<<(n + B - 1) / B, B, 0, stream>>>(dinv, n);
  k_count_deg<<<(E + B - 1) / B, B, 0, stream>>>(col, dinv, E);
  k_dinv<<<(n + B - 1) / B, B, 0, stream>>>(dinv, n);
  k_norm<<<(E + n + B - 1) / B, B, 0, stream>>>(row, col, dinv, norm, E, n);

  // 2) h0 = relu(x @ lin1_w + lin1_b)  — WMMA f32 16x16x4
  dim3 g1((n + 15) / 16, HID / 16);
  k_gemm1<<<g1, 32, 0, stream>>>(x, lin1_w, lin1_b, h0, n);

  // 3) GCNII layers
  const float* hcur = h0;
  for (int i = 0; i < nlayers; ++i) {
    float beta = logf(0.5f / (float)(i + 1) + 1.0f);
    k_init_s<<<(int)((nh + B - 1) / B), B, 0, stream>>>(h0, s, nh);
    long waves = (long)E + n;
    k_scatter<<<(int)((waves * 32 + B - 1) / B), B, 0, stream>>>(row, col, norm,
                                                                 hcur, s, E, n);
    k_gemm_layer<<<g1, 32, 0, stream>>>(s, convw + (long)i * HID * HID, h, beta, n);
    hcur = h;
  }

  // 4) global mean pool + final linear
  k_pool_zero<<<(NGRAPH * HID + NGRAPH + B - 1) / B, B, 0, stream>>>(pooled, counts);
  k_pool<<<(int)(((long)n * 32 + B - 1) / B), B, 0, stream>>>(batch, hcur, pooled,
                                                              counts, n);
  k_final<<<(NGRAPH * OUT_F + B - 1) / B, B, 0, stream>>>(pooled, counts, lin2_w,
                                                          lin2_b, out);
}